// LearnablePerm_71253507441027
// MI455X (gfx1250) — compile-verified
//
#include <hip/hip_runtime.h>
#include <math.h>

// LearnablePerm (block-diagonal hard permutation via Sinkhorn + Hungarian)
// MI455X / gfx1250: wave32, WMMA 16x16x4 f32 for Sinkhorn row/col sums,
// async global<->LDS staging (ASYNCcnt) for the 64KB tiles.

#define NB 64
#define BS 128
#define TAU_F 3.0f
#define SK_ITERS 5

typedef __attribute__((ext_vector_type(2))) float v2f;
typedef __attribute__((ext_vector_type(8))) float v8f;

#if __has_builtin(__builtin_amdgcn_wmma_f32_16x16x4_f32)
#define HAVE_WMMA_F32X4 1
#else
#define HAVE_WMMA_F32X4 0
#endif

// ---------------------------------------------------------------------------
// helpers
// ---------------------------------------------------------------------------
__device__ __forceinline__ float sel8(v8f c, int s) {
    float r = c[0];
    r = (s == 1) ? c[1] : r;
    r = (s == 2) ? c[2] : r;
    r = (s == 3) ? c[3] : r;
    r = (s == 4) ? c[4] : r;
    r = (s == 5) ? c[5] : r;
    r = (s == 6) ? c[6] : r;
    r = (s == 7) ? c[7] : r;
    return r;
}

// wave-relative LDS byte offset of a __shared__ location (generic-address
// low 32 bits == LDS aperture offset on gfx1250)
__device__ __forceinline__ unsigned lds_off(const void* p) {
    return (unsigned)(size_t)p;
}

// Row-normalize a BSxBS tile in LDS using one wave32.
// Row sums via V_WMMA_F32_16X16X4_F32: C += A(16x4 slice of P) * B(ones 4x16).
// Every column of C equals the row-sum vector; extraction uses only the
// documented C layout (lanes 0-15 -> rows v, lanes 16-31 -> rows v+8).
__device__ __forceinline__ void row_normalize(float* t, int lane) {
#if HAVE_WMMA_F32X4
    const int half = lane >> 4;           // which 16-lane half
    const int l15  = lane & 15;
    const int sel  = lane & 7;
    const int chalf = (lane >> 3) & 1;    // which 64-col half this lane divides
    v2f ones; ones[0] = 1.0f; ones[1] = 1.0f;
    for (int s8 = 0; s8 < 8; ++s8) {      // strips of 16 rows
        const int rbase = s8 * 16;
        const int arow  = rbase + l15;    // A layout: lanes 0-15 M (K,K+1), lanes 16-31 M (K+2,K+3)
        const int koff  = half * 2;
        v8f c = {};
        for (int kc = 0; kc < 32; ++kc) { // K = 128 in chunks of 4
            v2f a;
            a[0] = t[arow * BS + kc * 4 + koff];
            a[1] = t[arow * BS + kc * 4 + koff + 1];
            c = __builtin_amdgcn_wmma_f32_16x16x4_f32(
                    false, a, false, ones, (short)0, c, false, false);
        }
        // lane half h holds rowsums of rows rbase + h*8 + v in c[v]
        const float rs  = sel8(c, sel);
        const float inv = 1.0f / rs;
        const int   row = rbase + half * 8 + sel;
        const int   cb  = chalf * 64;     // two lanes per row split the columns
        for (int k = 0; k < 64; ++k) t[row * BS + cb + k] *= inv;
    }
#else
    for (int r = lane; r < BS; r += 32) {
        float s = 0.0f;
        for (int k = 0; k < BS; ++k) s += t[r * BS + k];
        const float inv = 1.0f / s;
        for (int k = 0; k < BS; ++k) t[r * BS + k] *= inv;
    }
#endif
}

// Col-normalize: column sums computed as row sums of the transpose, so only
// the documented A layout is needed (A loaded transposed from LDS), B = ones.
__device__ __forceinline__ void col_normalize(float* t, int lane) {
#if HAVE_WMMA_F32X4
    const int half = lane >> 4;
    const int l15  = lane & 15;
    const int sel  = lane & 7;
    const int rhalf = (lane >> 3) & 1;
    v2f ones; ones[0] = 1.0f; ones[1] = 1.0f;
    for (int s8 = 0; s8 < 8; ++s8) {      // strips of 16 columns
        const int nbase = s8 * 16;
        const int acol  = nbase + l15;
        const int koff  = half * 2;
        v8f c = {};
        for (int kc = 0; kc < 32; ++kc) { // K = 128 rows in chunks of 4
            v2f a;
            a[0] = t[(kc * 4 + koff) * BS + acol];
            a[1] = t[(kc * 4 + koff + 1) * BS + acol];
            c = __builtin_amdgcn_wmma_f32_16x16x4_f32(
                    false, a, false, ones, (short)0, c, false, false);
        }
        const float cs  = sel8(c, sel);
        const float inv = 1.0f / cs;
        const int   col = nbase + half * 8 + sel;
        const int   rb  = rhalf * 64;
        for (int r = 0; r < 64; ++r) t[(rb + r) * BS + col] *= inv;
    }
#else
    for (int cidx = lane; cidx < BS; cidx += 32) {
        float s = 0.0f;
        for (int r = 0; r < BS; ++r) s += t[r * BS + cidx];
        const float inv = 1.0f / s;
        for (int r = 0; r < BS; ++r) t[r * BS + cidx] *= inv;
    }
#endif
}

// ---------------------------------------------------------------------------
// Kernel 1: Sinkhorn. One wave32 per 128x128 block, tile resident in LDS.
// Tile staged in/out with gfx1250 async global<->LDS (no VGPR round-trip).
// ---------------------------------------------------------------------------
__global__ __launch_bounds__(32) void sinkhorn_kernel(
        const float* __restrict__ W, const float* __restrict__ scale_p,
        float* __restrict__ P) {
    __shared__ float t[BS * BS];          // 64 KB tile
    const int b = blockIdx.x;
    const int lane = threadIdx.x;
    const float scale = scale_p[0];
    const float* Wb = W + (size_t)b * BS * BS;

    // ---- async stage-in: lane l copies bytes [16l,16l+16) of each 512B stripe
    for (int i = lane * 4; i < BS * BS; i += 32 * 4) {
        const float* g = Wb + i;
        unsigned lo = lds_off(&t[i]);
        asm volatile("global_load_async_to_lds_b128 %0, %1, off"
                     :: "v"(lo), "v"(g) : "memory");
    }
    asm volatile("s_wait_asynccnt 0" ::: "memory");
    __syncthreads();

    // straight-through clamp, divide by tau (in place)
    for (int i = lane; i < BS * BS; i += 32) {
        float w = t[i] * scale;
        w = fminf(1.0f, fmaxf(-1.0f, w));
        t[i] = w * (1.0f / TAU_F);
    }
    __syncthreads();

    // row softmax: exp(x - rowmax), then row-normalize (the denominator)
    for (int r = lane; r < BS; r += 32) {
        float m = -INFINITY;
        for (int k = 0; k < BS; ++k) m = fmaxf(m, t[r * BS + k]);
        for (int k = 0; k < BS; ++k) t[r * BS + k] = expf(t[r * BS + k] - m);
    }
    __syncthreads();
    row_normalize(t, lane);
    __syncthreads();

    for (int it = 0; it < SK_ITERS; ++it) {
        row_normalize(t, lane);
        __syncthreads();
        col_normalize(t, lane);
        __syncthreads();
    }

    // ---- async stage-out of P (LDS -> global); DS writes must be complete
    asm volatile("s_wait_dscnt 0" ::: "memory");
    float* Pb = P + (size_t)b * BS * BS;
    for (int i = lane * 4; i < BS * BS; i += 32 * 4) {
        float* g = Pb + i;
        unsigned lo = lds_off(&t[i]);
        asm volatile("global_store_async_from_lds_b128 %0, %1, off"
                     :: "v"(g), "v"(lo) : "memory");
    }
    asm volatile("s_wait_asynccnt 0" ::: "memory");
}

// ---------------------------------------------------------------------------
// Kernel 2: Hungarian (Jonker-Volgenant) per block, one wave32 per block.
// Columns j = 1 + lane + 32q (q = 0..3) live per-lane; u/v/p/way in LDS.
// Cost = -P read from the L2-resident P buffer (prefetched).
// ---------------------------------------------------------------------------
__global__ __launch_bounds__(32) void hungarian_kernel(
        const float* __restrict__ P, int* __restrict__ cols) {
    const int b = blockIdx.x;
    const int lane = threadIdx.x;
    const float* Pb = P + (size_t)b * BS * BS;
    const double DINF = 1e18;

    // warm the block's cost tile into cache (gfx1250 global_prefetch path)
    for (int k = 0; k < 16; ++k)
        __builtin_prefetch(Pb + (size_t)(lane + k * 32) * 32, 0, 0);

    __shared__ double u[BS + 1], v[BS + 1];
    __shared__ int p[BS + 1], way[BS + 1];
    for (int j = lane; j <= BS; j += 32) { u[j] = 0.0; v[j] = 0.0; p[j] = 0; }
    __syncthreads();

    double minv[4];
    int used[4];

    for (int i = 1; i <= BS; ++i) {
        p[0] = i;                         // all lanes write identical value
        int j0 = 0;
#pragma unroll
        for (int q = 0; q < 4; ++q) { minv[q] = DINF; used[q] = 0; }

        while (true) {
            if (j0 > 0) {                 // mark used[j0] in its owner lane
                if (lane == ((j0 - 1) & 31)) used[(j0 - 1) >> 5] = 1;
            }
            const int i0 = p[j0];
            const double ui0 = u[i0];
#pragma unroll
            for (int q = 0; q < 4; ++q) {
                const int j = 1 + lane + 32 * q;
                if (!used[q]) {
                    const double cur =
                        (double)(-Pb[(i0 - 1) * BS + (j - 1)]) - ui0 - v[j];
                    if (cur < minv[q]) { minv[q] = cur; way[j] = j0; }
                }
            }
            // argmin over free columns, tie-break to smallest j (np.argmin)
            double best = DINF;
            int bestj = 0x7fffffff;
#pragma unroll
            for (int q = 0; q < 4; ++q) {
                const int j = 1 + lane + 32 * q;
                if (!used[q] &&
                    (minv[q] < best || (minv[q] == best && j < bestj))) {
                    best = minv[q]; bestj = j;
                }
            }
            for (int off = 16; off >= 1; off >>= 1) {
                const double ob = __shfl_xor(best, off, 32);
                const int oj = __shfl_xor(bestj, off, 32);
                if (ob < best || (ob == best && oj < bestj)) {
                    best = ob; bestj = oj;
                }
            }
            const double delta = best;
            const int j1 = bestj;

            // u[p[used]] += delta ; v[used] -= delta ; minv[free] -= delta
            if (lane == 0) u[i] += delta;         // j = 0 entry (p[0] == i)
#pragma unroll
            for (int q = 0; q < 4; ++q) {
                const int j = 1 + lane + 32 * q;
                if (used[q]) { u[p[j]] += delta; v[j] -= delta; }
                else         { minv[q] -= delta; }
            }
            j0 = j1;
            if (p[j0] == 0) break;
        }
        // augment along the alternating path (wave-uniform)
        while (j0 != 0) {
            const int jn = way[j0];
            p[j0] = p[jn];
            j0 = jn;
        }
        __syncthreads();
    }
#pragma unroll
    for (int q = 0; q < 4; ++q) {
        const int j = 1 + lane + 32 * q;
        cols[b * BS + (p[j] - 1)] = j - 1;        // col_of_row
    }
}

// ---------------------------------------------------------------------------
// Kernel 3: write the 8192x8192 output directly. One workgroup per row;
// 128-bit zero stores with the single 1.0 folded in (bandwidth-bound).
// ---------------------------------------------------------------------------
__global__ __launch_bounds__(256) void write_out_kernel(
        const int* __restrict__ cols, float* __restrict__ out) {
    const int row = blockIdx.x;                   // 0..8191
    const int cg = (row & ~127) + cols[row];      // global column of the 1.0
    float4* orow = (float4*)(out + (size_t)row * (NB * BS));
    for (int q4 = threadIdx.x; q4 < (NB * BS) / 4; q4 += blockDim.x) {
        const int base = q4 * 4;
        float4 val;
        val.x = (cg == base + 0) ? 1.0f : 0.0f;
        val.y = (cg == base + 1) ? 1.0f : 0.0f;
        val.z = (cg == base + 2) ? 1.0f : 0.0f;
        val.w = (cg == base + 3) ? 1.0f : 0.0f;
        orow[q4] = val;
    }
}

// ---------------------------------------------------------------------------
// launch
// ---------------------------------------------------------------------------
extern "C" void kernel_launch(void* const* d_in, const int* in_sizes, int n_in,
                              void* d_out, int out_size, void* d_ws,
                              size_t ws_size, hipStream_t stream) {
    (void)in_sizes; (void)n_in; (void)out_size; (void)ws_size;
    const float* W = (const float*)d_in[0];      // (64,128,128) f32
    const float* scale = (const float*)d_in[1];  // scalar f32
    float* out = (float*)d_out;                  // (8192,8192) f32

    // workspace layout: P (4 MB) then col_of_row (32 KB)
    float* Pbuf = (float*)d_ws;
    int* cols = (int*)((char*)d_ws + (size_t)NB * BS * BS * sizeof(float));

    sinkhorn_kernel<<<NB, 32, 0, stream>>>(W, scale, Pbuf);
    hungarian_kernel<<<NB, 32, 0, stream>>>(Pbuf, cols);
    write_out_kernel<<<NB * BS, 256, 0, stream>>>(cols, out);
}